// CGLayer_74363063763120
// MI455X (gfx1250) — compile-verified
//
#include <hip/hip_runtime.h>

#define NB   2
#define NN   256
#define TAUC 16

typedef float v2f __attribute__((ext_vector_type(2)));
typedef float v8f __attribute__((ext_vector_type(8)));

// ---- constant descriptor tables -------------------------------------------
// CG table offsets, indexed by (l1*3+l2)*3+l ; -1 = invalid triple
__device__ const int d_CG_OFF[27] = {
     0, -1, -1,   -1,  1, -1,   -1, -1, 10,
    -1, 35, -1,   44, 53, 80,   -1,125,170,
    -1, -1,245,   -1,270,315,  390,415,490 };
__device__ const int d_NBLK[3] = {3, 6, 6};
__device__ const int d_BL1[3][6] = { {0,1,2,0,0,0}, {0,1,1,1,2,2}, {0,1,1,2,2,2} };
__device__ const int d_BL2[3][6] = { {0,1,2,0,0,0}, {1,0,1,2,1,2}, {2,1,2,0,1,2} };
// global pair-counter base for T table per (l, block)
__device__ const int d_TBASE[3][6] = { {0,1,10,0,0,0}, {35,38,41,50,65,80}, {105,110,119,134,139,154} };
__device__ const int d_VOFF[3] = {0, 1, 4};          // row offset of l into 9-row layout
__device__ const int d_OUT_OFF[3] = {0, 8192, 32768};
__device__ const int d_OUT_LEN[3] = {8192, 24576, 40960};

// ---- Clebsch-Gordan (Racah) in device double ------------------------------
__device__ __forceinline__ double dfact(int n) {
  double r = 1.0;
  for (int i = 2; i <= n; ++i) r *= (double)i;
  return r;
}

__device__ float cg_coef(int j1, int j2, int j3, int m1, int m2, int m3) {
  if (m1 + m2 != m3) return 0.0f;
  double pre = sqrt((2.0 * j3 + 1.0) * dfact(j3 + j1 - j2) * dfact(j3 - j1 + j2) *
                    dfact(j1 + j2 - j3) / dfact(j1 + j2 + j3 + 1));
  pre *= sqrt(dfact(j3 + m3) * dfact(j3 - m3) * dfact(j1 - m1) * dfact(j1 + m1) *
              dfact(j2 - m2) * dfact(j2 + m2));
  int kmin = 0;
  if (j2 - j3 - m1 > kmin) kmin = j2 - j3 - m1;
  if (j1 - j3 + m2 > kmin) kmin = j1 - j3 + m2;
  int kmax = j1 + j2 - j3;
  if (j1 - m1 < kmax) kmax = j1 - m1;
  if (j2 + m2 < kmax) kmax = j2 + m2;
  double s = 0.0;
  for (int k = kmin; k <= kmax; ++k) {
    double den = dfact(k) * dfact(j1 + j2 - j3 - k) * dfact(j1 - m1 - k) *
                 dfact(j2 + m2 - k) * dfact(j3 - j2 + m1 + k) * dfact(j3 - j1 - m2 + k);
    s += ((k & 1) ? -1.0 : 1.0) / den;
  }
  return (float)(pre * s);
}

// K0: fill CG table (615 floats) in workspace
__global__ void k_init_cg(float* __restrict__ cg) {
  for (int l1 = 0; l1 < 3; ++l1)
    for (int l2 = 0; l2 < 3; ++l2)
      for (int l = 0; l < 3; ++l) {
        int off = d_CG_OFF[(l1 * 3 + l2) * 3 + l];
        if (off < 0) continue;
        int n2 = 2 * l2 + 1, nk = 2 * l + 1;
        int total = (2 * l1 + 1) * n2 * nk;
        for (int e = threadIdx.x; e < total; e += blockDim.x) {
          int ki = e % nk;
          int r = e / nk;
          int m2i = r % n2;
          int m1i = r / n2;
          cg[off + e] = cg_coef(l1, l2, l, m1i - l1, m2i - l2, ki - l);
        }
      }
}

// K1: conn -> float, and pack v0/v1/v2 into [node][144] (rows l0m0,l1m0..2,l2m0..4 x 16ch)
__global__ void k_pack(const int* __restrict__ conn, const float* __restrict__ v0,
                       const float* __restrict__ v1, const float* __restrict__ v2,
                       float* __restrict__ connf, float* __restrict__ vcat) {
  int gid = blockIdx.x * blockDim.x + threadIdx.x;
  const int nconn = NB * NN * NN;      // 131072
  const int nvcat = NB * NN * 144;     // 73728
  if (gid < nconn) connf[gid] = (float)conn[gid];
  int g2 = gid - nconn;
  if (g2 >= 0 && g2 < nvcat) {
    int node = g2 / 144, col = g2 % 144;
    float val;
    if (col < 16)      val = v0[node * 16 + col];
    else if (col < 64) val = v1[node * 48 + (col - 16)];
    else               val = v2[node * 80 + (col - 64)];
    vcat[g2] = val;
  }
}

// K2: vmp[b] = connf[b](256x256) x vcat[b](256x144) via V_WMMA_F32_16X16X4_F32
// One wave per 16x16 output tile. EXEC all-ones (no divergence before WMMA).
__global__ void k_mp_wmma(const float* __restrict__ connf, const float* __restrict__ vcat,
                          float* __restrict__ vmp) {
  const int mt = blockIdx.x;      // 16 M-tiles
  const int nt = blockIdx.y;      // 9  N-tiles
  const int b  = blockIdx.z;      // batch
  const int lane = threadIdx.x;   // 0..31
  const int half = lane >> 4;
  const int l16  = lane & 15;
  const float* A  = connf + (size_t)b * NN * NN;
  const float* Bm = vcat  + (size_t)b * NN * 144;
  const int arow = mt * 16 + l16;
  const int bcol = nt * 16 + l16;
  v8f acc = {};
  for (int k0 = 0; k0 < NN; k0 += 4) {
    int k = k0 + half * 2;
    // A 16x4 fragment: lanes 0-15 hold K=k0,k0+1 ; lanes 16-31 hold K=k0+2,k0+3
    v2f a;
    a.x = A[arow * NN + k];
    a.y = A[arow * NN + k + 1];
    // B 4x16 fragment, mirrored K distribution
    v2f bb;
    bb.x = Bm[(size_t)k * 144 + bcol];
    bb.y = Bm[(size_t)(k + 1) * 144 + bcol];
    acc = __builtin_amdgcn_wmma_f32_16x16x4_f32(false, a, false, bb, (short)0, acc,
                                                false, false);
  }
  float* Out = vmp + (size_t)b * NN * 144;
  for (int r = 0; r < 8; ++r) {
    int row = mt * 16 + half * 8 + r;   // C/D layout: VGPR r holds M=r (lanes0-15), M=r+8 (lanes16-31)
    Out[(size_t)row * 144 + bcol] = acc[r];
  }
}

// K3: Ssum[node][9] = sum_j s_l[b,i,j,m] ; coalesced strided reads, fixed-order LDS tree
__global__ void k_ssum(const float* __restrict__ s0, const float* __restrict__ s1,
                       const float* __restrict__ s2, float* __restrict__ ssum) {
  int node = blockIdx.x;      // 512
  int tid = threadIdx.x;      // 128
  __shared__ float red[128];
  const float* sp[3] = {s0, s1, s2};
  for (int l = 0; l < 3; ++l) {
    int md = 2 * l + 1, len = NN * md;
    const float* base = sp[l] + (size_t)node * len;
    float acc[5] = {0.f, 0.f, 0.f, 0.f, 0.f};
    for (int e = tid; e < len; e += 128) acc[e % md] += base[e];
    for (int m = 0; m < md; ++m) {
      red[tid] = acc[m];
      __syncthreads();
      for (int s = 64; s > 0; s >>= 1) {
        if (tid < s) red[tid] += red[tid + s];
        __syncthreads();
      }
      if (tid == 0) ssum[node * 16 + d_VOFF[l] + m] = red[0];
      __syncthreads();
    }
  }
}

// K4: per-node CG products + channel mixing + j-factored rel stage -> unnormalized out
__global__ void k_node(const float* __restrict__ vmp, const float* __restrict__ ssum,
                       const float* __restrict__ cg,
                       const float* __restrict__ wnl0, const float* __restrict__ wnl1,
                       const float* __restrict__ wnl2,
                       const float* __restrict__ wrel0, const float* __restrict__ wrel1,
                       const float* __restrict__ wrel2, float* __restrict__ out) {
  int node = blockIdx.x;     // 512
  int tid = threadIdx.x;     // 64
  __shared__ float s_v[144];
  __shared__ float s_ss[16];
  __shared__ float s_T[179 * 16];
  __shared__ float s_mnl[144];
  for (int i = tid; i < 144; i += 64) s_v[i] = vmp[(size_t)node * 144 + i];
  if (tid < 16) s_ss[tid] = ssum[node * 16 + tid];
  __syncthreads();
  const float* wnl[3]  = {wnl0, wnl1, wnl2};
  const float* wrel[3] = {wrel0, wrel1, wrel2};
  const int t = tid & 15;     // output channel
  const int grp = tid >> 4;   // 4 work groups

  // Stage A1: T[pair][t] = sum_{c,d} vmp[l1][m1][c] * vmp[l2][m2][d] * wnl[l][bi*256+c*16+d][t]
  int p = 0;
  for (int l = 0; l < 3; ++l) {
    for (int bi = 0; bi < d_NBLK[l]; ++bi) {
      int l1 = d_BL1[l][bi], l2 = d_BL2[l][bi];
      int n1 = 2 * l1 + 1, n2 = 2 * l2 + 1;
      const float* W = wnl[l] + (size_t)bi * 256 * 16;
      for (int m1i = 0; m1i < n1; ++m1i)
        for (int m2i = 0; m2i < n2; ++m2i) {
          if ((p & 3) == grp) {
            const float* a  = s_v + (d_VOFF[l1] + m1i) * 16;
            const float* bv = s_v + (d_VOFF[l2] + m2i) * 16;
            float T = 0.f;
            for (int c = 0; c < 16; ++c) {
              const float* Wc = W + (size_t)(c * 16) * 16 + t;
              float tt = 0.f;
              for (int d = 0; d < 16; ++d) tt += bv[d] * Wc[(size_t)d * 16];
              T += a[c] * tt;
            }
            s_T[p * 16 + t] = T;
          }
          ++p;
        }
    }
  }
  __syncthreads();

  // Stage A2: mixed_nl[l][k][t] = sum_{blocks,(m1,m2)} C[m1,m2,k] * T[pair][t]
  for (int row = grp; row < 9; row += 4) {
    int l = (row == 0) ? 0 : (row < 4 ? 1 : 2);
    int k = row - d_VOFF[l];
    float acc = 0.f;
    for (int bi = 0; bi < d_NBLK[l]; ++bi) {
      int l1 = d_BL1[l][bi], l2 = d_BL2[l][bi];
      int n2 = 2 * l2 + 1, nk = 2 * l + 1;
      const float* C = cg + d_CG_OFF[(l1 * 3 + l2) * 3 + l];
      int base = d_TBASE[l][bi];
      int np = (2 * l1 + 1) * n2;
      for (int pp = 0; pp < np; ++pp) {
        float c = C[pp * nk + k];
        if (c != 0.0f) acc += c * s_T[(base + pp) * 16 + t];
      }
    }
    s_mnl[row * 16 + t] = acc;
  }
  __syncthreads();

  // Stage B (j-sum factored): out[l][k][t] =
  //   sum_{blocks,m1} (sum_{m2} C[m1,m2,k]*Ssum[l2][m2]) * (sum_c mnl[l1][m1][c]*wrel[l][bi*16+c][t])
  for (int row = grp; row < 9; row += 4) {
    int l = (row == 0) ? 0 : (row < 4 ? 1 : 2);
    int k = row - d_VOFF[l];
    float acc = 0.f;
    for (int bi = 0; bi < d_NBLK[l]; ++bi) {
      int l1 = d_BL1[l][bi], l2 = d_BL2[l][bi];
      int n1 = 2 * l1 + 1, n2 = 2 * l2 + 1, nk = 2 * l + 1;
      const float* C = cg + d_CG_OFF[(l1 * 3 + l2) * 3 + l];
      const float* Wr = wrel[l] + (size_t)bi * 16 * 16;
      for (int m1i = 0; m1i < n1; ++m1i) {
        float R = 0.f;
        for (int c = 0; c < 16; ++c) R += s_mnl[(d_VOFF[l1] + m1i) * 16 + c] * Wr[c * 16 + t];
        float g = 0.f;
        for (int m2i = 0; m2i < n2; ++m2i) {
          float cc = C[(m1i * n2 + m2i) * nk + k];
          g += cc * s_ss[d_VOFF[l2] + m2i];
        }
        acc += g * R;
      }
    }
    out[d_OUT_OFF[l] + ((size_t)node * (2 * l + 1) + k) * 16 + t] = acc;
  }
}

// K5: deterministic per-part partial sums (64 blocks per part)
__global__ void k_redpart(const float* __restrict__ out, float* __restrict__ red) {
  int part = blockIdx.x >> 6;
  int blk = blockIdx.x & 63;
  int tid = threadIdx.x;   // 256
  __shared__ float sm[256];
  int off = d_OUT_OFF[part], len = d_OUT_LEN[part];
  float acc = 0.f;
  for (int e = blk * 256 + tid; e < len; e += 64 * 256) acc += out[off + e];
  sm[tid] = acc;
  __syncthreads();
  for (int s = 128; s > 0; s >>= 1) {
    if (tid < s) sm[tid] += sm[tid + s];
    __syncthreads();
  }
  if (tid == 0) red[part * 64 + blk] = sm[0];
}

// K6: fixed-order final reduction per part
__global__ void k_redfin(const float* __restrict__ red, float* __restrict__ tot) {
  int part = threadIdx.x;
  if (part < 3) {
    float a = 0.f;
    for (int i = 0; i < 64; ++i) a += red[part * 64 + i];
    tot[part] = a;
  }
}

// K7: in-place normalization out[i] /= tot[part(i)]
__global__ void k_scale(float* __restrict__ out, const float* __restrict__ tot) {
  int gid = blockIdx.x * blockDim.x + threadIdx.x;
  if (gid >= 73728) return;
  int part = (gid < 8192) ? 0 : (gid < 32768 ? 1 : 2);
  out[gid] = out[gid] / tot[part];
}

extern "C" void kernel_launch(void* const* d_in, const int* in_sizes, int n_in,
                              void* d_out, int out_size, void* d_ws, size_t ws_size,
                              hipStream_t stream) {
  (void)in_sizes; (void)n_in; (void)out_size; (void)ws_size;
  // setup_inputs() dict order: v0,s0,wnl0,wrel0, v1,s1,wnl1,wrel1, v2,s2,wnl2,wrel2, conn
  const float* v0    = (const float*)d_in[0];
  const float* s0    = (const float*)d_in[1];
  const float* wnl0  = (const float*)d_in[2];
  const float* wrel0 = (const float*)d_in[3];
  const float* v1    = (const float*)d_in[4];
  const float* s1    = (const float*)d_in[5];
  const float* wnl1  = (const float*)d_in[6];
  const float* wrel1 = (const float*)d_in[7];
  const float* v2    = (const float*)d_in[8];
  const float* s2    = (const float*)d_in[9];
  const float* wnl2  = (const float*)d_in[10];
  const float* wrel2 = (const float*)d_in[11];
  const int*   conn  = (const int*)d_in[12];
  float* out = (float*)d_out;
  float* ws  = (float*)d_ws;

  // workspace layout (floats): ~1.15 MB total
  float* wsCG    = ws;                       // 615 (pad 640)
  float* wsCONNF = ws + 640;                 // 131072
  float* wsVCAT  = wsCONNF + 131072;         // 73728
  float* wsVMP   = wsVCAT + 73728;           // 73728
  float* wsSSUM  = wsVMP + 73728;            // 8192
  float* wsRED   = wsSSUM + 8192;            // 192
  float* wsTOT   = wsRED + 192;              // 3

  k_init_cg<<<1, 128, 0, stream>>>(wsCG);
  k_pack<<<(131072 + 73728) / 256, 256, 0, stream>>>(conn, v0, v1, v2, wsCONNF, wsVCAT);
  k_mp_wmma<<<dim3(16, 9, NB), 32, 0, stream>>>(wsCONNF, wsVCAT, wsVMP);
  k_ssum<<<NB * NN, 128, 0, stream>>>(s0, s1, s2, wsSSUM);
  k_node<<<NB * NN, 64, 0, stream>>>(wsVMP, wsSSUM, wsCG, wnl0, wnl1, wnl2,
                                     wrel0, wrel1, wrel2, out);
  k_redpart<<<192, 256, 0, stream>>>(out, wsRED);
  k_redfin<<<1, 32, 0, stream>>>(wsRED, wsTOT);
  k_scale<<<(73728 + 255) / 256, 256, 0, stream>>>(out, wsTOT);
}